// KmaxLayer_21818433864134
// MI455X (gfx1250) — compile-verified
//
#include <hip/hip_runtime.h>
#include <stdint.h>

// Top-k (k=16) threshold + masked normalization over rows of 1024 fp32.
// One wave32 per row; async global->LDS double buffering (CDNA5 ASYNCcnt path).

#define K_TOP     16
#define M_ROW     1024
#define WAVES_PB  4
#define THREADS_PB (WAVES_PB * 32)
#define CHUNKS    8   // 8 chunks * 128 floats (512B) = 1024 floats per row

typedef int vi4 __attribute__((vector_size(4 * sizeof(int))));
typedef __attribute__((address_space(1))) vi4 GlobalVi4;
typedef __attribute__((address_space(3))) vi4 LdsVi4;
typedef __attribute__((address_space(3))) void LdsVoid;

// 16B async copy: global -> LDS, tracked by ASYNCcnt (no VGPR destination).
__device__ __forceinline__ void async_copy16(const float* g, float* l) {
#if __has_builtin(__builtin_amdgcn_global_load_async_to_lds_b128)
  __builtin_amdgcn_global_load_async_to_lds_b128((GlobalVi4*)g, (LdsVi4*)l, 0, 0);
#else
  asm volatile("global_load_async_to_lds_b128 %0, %1, off"
               :
               : "v"((unsigned)(uintptr_t)(LdsVoid*)l),
                 "v"((unsigned long long)(uintptr_t)g)
               : "memory");
#endif
}

__device__ __forceinline__ void wait_async(int next_issued) {
#if __has_builtin(__builtin_amdgcn_s_wait_asynccnt)
  if (next_issued) __builtin_amdgcn_s_wait_asynccnt(CHUNKS);
  else             __builtin_amdgcn_s_wait_asynccnt(0);
#else
  if (next_issued) asm volatile("s_wait_asynccnt 0x8" ::: "memory");
  else             asm volatile("s_wait_asynccnt 0x0" ::: "memory");
#endif
}

__global__ __launch_bounds__(THREADS_PB)
void KmaxLayer_topk_norm_kernel(const float* __restrict__ in,
                                float* __restrict__ out,
                                int nrows) {
  __shared__ __align__(16) float buf[WAVES_PB][2][M_ROW];

  const int lane = threadIdx.x & 31;
  const int wv   = threadIdx.x >> 5;
  const int gw   = blockIdx.x * WAVES_PB + wv;
  const int nw   = gridDim.x * WAVES_PB;

  int row = gw;
  if (row < nrows) {
    const float* src = in + (size_t)row * M_ROW;
#pragma unroll
    for (int c = 0; c < CHUNKS; ++c)
      async_copy16(src + c * 128 + lane * 4, &buf[wv][0][c * 128 + lane * 4]);
  }

  int sel = 0;
  for (; row < nrows; row += nw) {
    const int nxt = row + nw;
    if (nxt < nrows) {
      const float* src = in + (size_t)nxt * M_ROW;
#pragma unroll
      for (int c = 0; c < CHUNKS; ++c)
        async_copy16(src + c * 128 + lane * 4,
                     &buf[wv][sel ^ 1][c * 128 + lane * 4]);
      wait_async(1);   // ASYNCcnt <= 8: current buffer's 8 loads done (in-order)
    } else {
      wait_async(0);   // drain: last buffer fully landed
    }

    // Pull this lane's 32 elements (stride-512B float4 chunks) into registers.
    float v[32];
#pragma unroll
    for (int c = 0; c < CHUNKS; ++c) {
      const float4 q =
          *(const float4*)&buf[wv][sel][c * 128 + lane * 4];
      v[c * 4 + 0] = q.x;
      v[c * 4 + 1] = q.y;
      v[c * 4 + 2] = q.z;
      v[c * 4 + 3] = q.w;
    }

    // --- selection: find 16th-largest value of the row (duplicate-safe) ---
    float prev = __builtin_inff();
    float kth  = -__builtin_inff();
    int removed = 0;
#pragma unroll 1
    for (int it = 0; it < K_TOP; ++it) {
      // wave-wide max over elements strictly below previous distinct max
      float lm = -__builtin_inff();
#pragma unroll
      for (int j = 0; j < 32; ++j) {
        const float x = v[j];
        lm = fmaxf(lm, (x < prev) ? x : -__builtin_inff());
      }
#pragma unroll
      for (int off = 16; off > 0; off >>= 1)
        lm = fmaxf(lm, __shfl_xor(lm, off, 32));

      // count occurrences of this distinct max across the wave
      int c = 0;
#pragma unroll
      for (int j = 0; j < 32; ++j) c += (v[j] == lm) ? 1 : 0;
#pragma unroll
      for (int off = 16; off > 0; off >>= 1)
        c += __shfl_xor(c, off, 32);

      removed += c;
      kth  = lm;
      prev = lm;
      if (removed >= K_TOP) break;
    }

    // --- masked sum + normalize ---
    float s = 0.0f;
#pragma unroll
    for (int j = 0; j < 32; ++j) s += (v[j] >= kth) ? v[j] : 0.0f;
#pragma unroll
    for (int off = 16; off > 0; off >>= 1) s += __shfl_xor(s, off, 32);
    const float inv = 1.0f / s;

    float* dst = out + (size_t)row * M_ROW;
#pragma unroll
    for (int c = 0; c < CHUNKS; ++c) {
      float4 o;
      o.x = (v[c * 4 + 0] >= kth) ? v[c * 4 + 0] * inv : 0.0f;
      o.y = (v[c * 4 + 1] >= kth) ? v[c * 4 + 1] * inv : 0.0f;
      o.z = (v[c * 4 + 2] >= kth) ? v[c * 4 + 2] * inv : 0.0f;
      o.w = (v[c * 4 + 3] >= kth) ? v[c * 4 + 3] * inv : 0.0f;
      *(float4*)(dst + c * 128 + lane * 4) = o;
    }

    sel ^= 1;
  }
}

extern "C" void kernel_launch(void* const* d_in, const int* in_sizes, int n_in,
                              void* d_out, int out_size, void* d_ws, size_t ws_size,
                              hipStream_t stream) {
  (void)n_in; (void)d_ws; (void)ws_size; (void)out_size;
  const float* in = (const float*)d_in[0];
  float* out = (float*)d_out;

  const int nrows = in_sizes[0] / M_ROW;          // 65536 for 16x4096x1024
  int blocks = (nrows + WAVES_PB - 1) / WAVES_PB; // 1 row per wave max
  if (blocks > 2048) blocks = 2048;               // ~8 rows pipelined per wave

  KmaxLayer_topk_norm_kernel<<<dim3(blocks), dim3(THREADS_PB), 0, stream>>>(
      in, out, nrows);
}